// BahdanauAttention_55164559950036
// MI455X (gfx1250) — compile-verified
//
#include <hip/hip_runtime.h>

// Bahdanau attention, B=65536, W=20, E=128, all f32.
// HBM-bound (~720 MB moved, ~31 us floor at 23.3 TB/s). No GEMM structure
// (tanh inside the e-reduction; context matvec couples both operands to b),
// so WMMA is inapplicable. CDNA5 paths used: async global->LDS tile
// streaming (ASYNCcnt) with double buffering + NT temporal hint on the
// streamed value tensor, wave32 shuffle reductions, hardware v_tanh_f32.

#define WDIM 20
#define EDIM 128
#define WAVES_PER_BLOCK 4
#define TILE_BYTES (WDIM * EDIM * 4)     // 10240 B per value tile
#define W2_OFF 0
#define W3_OFF TILE_BYTES
#define BUFS_OFF (2 * TILE_BYTES)        // 20480: start of per-wave buffers

// gfx12x CPol: TH in [2:0]; TH_NT = 1 (non-temporal). Hint only.
#define CPOL_RT 0
#define CPOL_NT 1

typedef int v4i __attribute__((ext_vector_type(4)));
typedef __attribute__((address_space(1))) v4i glb_v4i_t;   // global (AS1)
typedef __attribute__((address_space(3))) v4i lds_v4i_t;   // LDS (AS3)

// --- async 16B global->LDS copy (per lane) -------------------------------
template <int CPOL>
__device__ __forceinline__ void async_cp16(const void* gp, unsigned lds_off) {
#if __has_builtin(__builtin_amdgcn_global_load_async_to_lds_b128)
  __builtin_amdgcn_global_load_async_to_lds_b128(
      (glb_v4i_t*)(unsigned long long)gp,
      (lds_v4i_t*)lds_off,
      0, CPOL);
#else
  asm volatile("global_load_async_to_lds_b128 %0, %1, off"
               :: "v"(lds_off), "v"((unsigned long long)gp)
               : "memory");
#endif
}

#if __has_builtin(__builtin_amdgcn_s_wait_asynccnt)
#define WAIT_ASYNC(N) __builtin_amdgcn_s_wait_asynccnt(N)
#else
#define WAIT_ASYNC(N) asm volatile("s_wait_asynccnt %0" :: "i"(N) : "memory")
#endif

__device__ __forceinline__ float ftanh(float x) {
#if __has_builtin(__builtin_amdgcn_tanhf)
  return __builtin_amdgcn_tanhf(x);          // v_tanh_f32 (gfx1250 TRANS op)
#elif __has_builtin(__builtin_amdgcn_tanh_f32)
  return __builtin_amdgcn_tanh_f32(x);
#else
  return tanhf(x);
#endif
}

extern "C" __global__ __launch_bounds__(WAVES_PER_BLOCK * 32, 1)
void bahdanau_attn_kernel(const float* __restrict__ query,
                          const float* __restrict__ value,
                          const float* __restrict__ W1,
                          const float* __restrict__ W2,
                          const float* __restrict__ W3,
                          float* __restrict__ out,
                          int Btot) {
  extern __shared__ char smem[];  // [W2 10K][W3 10K][wave0 buf0|buf1]...[wave3 buf0|buf1]
  const int lane = threadIdx.x & 31;
  const int wave = threadIdx.x >> 5;

  // ---- stage W2, W3 into LDS once per block (async, two uniform loops) ---
  for (int i = threadIdx.x; i < (WDIM * EDIM) / 4; i += blockDim.x)
    async_cp16<CPOL_RT>(W2 + (size_t)i * 4, (unsigned)(W2_OFF + i * 16));
  for (int i = threadIdx.x; i < (WDIM * EDIM) / 4; i += blockDim.x)
    async_cp16<CPOL_RT>(W3 + (size_t)i * 4, (unsigned)(W3_OFF + i * 16));
  WAIT_ASYNC(0);
  __syncthreads();

  const int nwaves = gridDim.x * (blockDim.x >> 5);
  const int gwave  = blockIdx.x * (blockDim.x >> 5) + wave;
  const unsigned mybuf = BUFS_OFF + wave * 2 * TILE_BYTES;

  const float4 w1 = *(const float4*)(W1 + lane * 4);

  // ---- prefetch first tile (NT: value is streamed once, 671 MB >> L2) ----
  if (gwave < Btot) {
    const char* g = (const char*)value + (size_t)gwave * TILE_BYTES + lane * 16;
    const unsigned l = mybuf + lane * 16;
#pragma unroll
    for (int w = 0; w < WDIM; ++w) async_cp16<CPOL_NT>(g + w * 512, l + w * 512);
  }

  int parity = 0;
  for (int b = gwave; b < Btot; b += nwaves) {
    const int bn = b + nwaves;
    if (bn < Btot) {
      // prefetch next tile into other buffer, then wait for current tile
      const char* g = (const char*)value + (size_t)bn * TILE_BYTES + lane * 16;
      const unsigned l = mybuf + (unsigned)((parity ^ 1) * TILE_BYTES) + lane * 16;
#pragma unroll
      for (int w = 0; w < WDIM; ++w) async_cp16<CPOL_NT>(g + w * 512, l + w * 512);
      WAIT_ASYNC(WDIM);   // async loads complete in order: current tile done
    } else {
      WAIT_ASYNC(0);
    }

    const unsigned buf = mybuf + (unsigned)(parity * TILE_BYTES);

    const float4 q = *(const float4*)(query + (size_t)b * EDIM + lane * 4);
    const float rqx = q.x * w1.x, rqy = q.y * w1.y;
    const float rqz = q.z * w1.z, rqw = q.w * w1.w;

    // ---- scores: sc[w] = sum_e tanh(rq*v*W2) * W3 ------------------------
    float sc[WDIM];
#pragma unroll
    for (int w = 0; w < WDIM; ++w) {
      const float4 v  = *(const float4*)(smem + buf + w * 512 + lane * 16);
      const float4 c2 = *(const float4*)(smem + W2_OFF + w * 512 + lane * 16);
      const float4 c3 = *(const float4*)(smem + W3_OFF + w * 512 + lane * 16);
      const float sx = ftanh(rqx * v.x * c2.x);
      const float sy = ftanh(rqy * v.y * c2.y);
      const float sz = ftanh(rqz * v.z * c2.z);
      const float sw = ftanh(rqw * v.w * c2.w);
      sc[w] = sx * c3.x + sy * c3.y + sz * c3.z + sw * c3.w;
    }

    // ---- wave32 butterfly reductions over E ------------------------------
#pragma unroll
    for (int w = 0; w < WDIM; ++w) {
      float r = sc[w];
      r += __shfl_xor(r, 16, 32);
      r += __shfl_xor(r, 8, 32);
      r += __shfl_xor(r, 4, 32);
      r += __shfl_xor(r, 2, 32);
      r += __shfl_xor(r, 1, 32);
      sc[w] = r;   // broadcast to all lanes
    }

    // ---- softmax over W=20 (replicated per lane, cheap) ------------------
    float m = sc[0];
#pragma unroll
    for (int w = 1; w < WDIM; ++w) m = fmaxf(m, sc[w]);
    float s = 0.f;
#pragma unroll
    for (int w = 0; w < WDIM; ++w) { sc[w] = __expf(sc[w] - m); s += sc[w]; }
    const float inv = 1.0f / s;

    // ---- context = sum_w a[w] * value[b,w,:] (re-read tile from LDS) -----
    float ax = 0.f, ay = 0.f, az = 0.f, aw = 0.f;
#pragma unroll
    for (int w = 0; w < WDIM; ++w) {
      const float4 v = *(const float4*)(smem + buf + w * 512 + lane * 16);
      const float a = sc[w] * inv;
      ax += a * v.x; ay += a * v.y; az += a * v.z; aw += a * v.w;
    }
    float4 o; o.x = ax; o.y = ay; o.z = az; o.w = aw;
    *(float4*)(out + (size_t)b * EDIM + lane * 4) = o;

    parity ^= 1;
  }
}

extern "C" void kernel_launch(void* const* d_in, const int* in_sizes, int n_in,
                              void* d_out, int out_size, void* d_ws, size_t ws_size,
                              hipStream_t stream) {
  const float* query = (const float*)d_in[0];
  const float* value = (const float*)d_in[1];
  const float* W1p   = (const float*)d_in[2];
  const float* W2p   = (const float*)d_in[3];
  const float* W3p   = (const float*)d_in[4];
  float* out = (float*)d_out;

  const int Btot = in_sizes[0] / EDIM;  // 65536

  // 20 KB weights + 4 waves * 2 * 10 KB double buffers = 100 KB dynamic LDS
  // -> 3 blocks (12 waves) per 320 KB WGP; ~120 KB of value bytes in flight
  //    per WGP, ample to cover HBM latency at 23.3 TB/s.
  const size_t shmem = (size_t)BUFS_OFF + (size_t)WAVES_PER_BLOCK * 2 * TILE_BYTES;

  // 2048 blocks * 4 waves = 8192 waves; each processes 8 batch elements,
  // so the double-buffered pipeline amortizes well.
  dim3 grid(2048), block(WAVES_PER_BLOCK * 32);
  bahdanau_attn_kernel<<<grid, block, shmem, stream>>>(query, value, W1p, W2p, W3p,
                                                       out, Btot);
}